// SkipGramHierarchicalSoftmaxModel_63488206569708
// MI455X (gfx1250) — compile-verified
//
#include <hip/hip_runtime.h>
#include <math.h>

typedef float v2f __attribute__((ext_vector_type(2)));
typedef float v8f __attribute__((ext_vector_type(8)));

#define B_SZ   1024
#define C_SZ   10
#define L_SZ   18
#define CL     (C_SZ * L_SZ)   // 180 gathered logits per batch row
#define D_SZ   128             // embedding dim
#define NTILES 12              // ceil(180 / 16)
#define WAVES  4               // 128 threads/block

// One block per batch element. Fuses: emb gather -> 180 gathered dot products
// (via WMMA f32 16x16x4) -> +bias -> BCE-with-logits terms -> block partial sum.
__global__ __launch_bounds__(128) void hsm_fused_loss_kernel(
    const int*   __restrict__ input_nodes,  // [B]
    const int*   __restrict__ path_nodes,   // [B, C*L]
    const float* __restrict__ tree_codes,   // [B, C*L]
    const float* __restrict__ emb,          // [N, D]
    const float* __restrict__ W,            // [2N, D]
    const float* __restrict__ bias,         // [2N]
    float*       __restrict__ partial)      // [B] block partial sums
{
    __shared__ float xsh[D_SZ];
    __shared__ float wsum[WAVES];

    const int b    = blockIdx.x;
    const int tid  = threadIdx.x;
    const int lane = tid & 31;
    const int wave = tid >> 5;

    // Stage x = emb[input_nodes[b]] into LDS (128 floats, one per thread).
    const long long node = (long long)input_nodes[b];
    xsh[tid] = emb[node * D_SZ + tid];
    __syncthreads();

    // WMMA f32 16x16x4 operand mapping (wave32):
    //  A (16x4): VGPR0 = K = 0 (lanes 0-15) / 2 (lanes 16-31); VGPR1 = K = 1 / 3
    //  B (4x16): lane%16 = N column; same K split as A across wave halves.
    const int  half  = lane >> 4;       // 0 or 1
    const int  cofs  = half << 1;       // K sub-offset: 0 or 2
    const int  nlane = lane & 15;       // output column within the tile

    float lsum = 0.0f;

    #pragma unroll
    for (int t = 0; t < NTILES / WAVES; ++t) {
        const int  tile  = wave + t * WAVES;       // uniform per wave -> EXEC stays full
        const int  j     = tile * 16 + nlane;      // gathered-logit index in [0,192)
        const bool valid = (j < CL);
        const int  jsafe = valid ? j : 0;
        const int  row   = path_nodes[b * CL + jsafe];   // in [0, 2N)
        const float* wrow = W + (long long)row * D_SZ;

        v8f acc = {0.f, 0.f, 0.f, 0.f, 0.f, 0.f, 0.f, 0.f};

        #pragma unroll
        for (int k0 = 0; k0 < D_SZ; k0 += 4) {
            // A chunk: broadcast x[k0+cofs .. +1] from LDS (b64, same addr per half-wave)
            v2f a;
            a[0] = xsh[k0 + cofs];
            a[1] = xsh[k0 + cofs + 1];
            // B chunk: per-lane 8B contiguous load of the gathered W row (b64)
            const float2 wp = *(const float2*)(wrow + k0 + cofs);
            v2f bm;
            bm[0] = wp.x;
            bm[1] = wp.y;
            // acc = A * B + acc  (D[m][n] = dot(x, Wrow_n) for every m)
            acc = __builtin_amdgcn_wmma_f32_16x16x4_f32(
                false, a, false, bm, (short)0, acc, false, false);
        }

        // D row M=0 lives in acc[0] on lanes 0..15 (N = lane).
        const float z     = acc[0] + bias[row];
        const float tcode = tree_codes[b * CL + jsafe];
        // Numerically stable BCE-with-logits term.
        const float term  = fmaxf(z, 0.0f) - z * tcode + log1pf(__expf(-fabsf(z)));
        // Branchless mask: only lanes 0-15 with j<180 contribute (keeps EXEC full).
        const float mask  = (half == 0 && valid) ? 1.0f : 0.0f;
        lsum += mask * term;
    }

    // Deterministic wave reduction (lanes >=16 carry 0 already).
    #pragma unroll
    for (int off = 16; off > 0; off >>= 1)
        lsum += __shfl_down(lsum, off, 32);
    if (lane == 0) wsum[wave] = lsum;
    __syncthreads();

    if (tid == 0) {
        float s = 0.0f;
        #pragma unroll
        for (int wv = 0; wv < WAVES; ++wv) s += wsum[wv];
        partial[b] = s;
    }
}

// Single-block deterministic final reduction: mean over B*C*L terms.
__global__ __launch_bounds__(256) void hsm_reduce_kernel(
    const float* __restrict__ partial,  // [B]
    float*       __restrict__ out)      // [1]
{
    __shared__ float sh[256];
    const int tid = threadIdx.x;

    float s = 0.0f;
    for (int i = tid; i < B_SZ; i += 256)   // fixed traversal order
        s += partial[i];
    sh[tid] = s;
    __syncthreads();

    #pragma unroll
    for (int off = 128; off > 0; off >>= 1) {
        if (tid < off) sh[tid] += sh[tid + off];
        __syncthreads();
    }
    if (tid == 0)
        out[0] = sh[0] * (1.0f / (float)(B_SZ * CL));
}

extern "C" void kernel_launch(void* const* d_in, const int* in_sizes, int n_in,
                              void* d_out, int out_size, void* d_ws, size_t ws_size,
                              hipStream_t stream) {
    const int*   input_nodes = (const int*)  d_in[0];  // [B] int32
    const int*   path_nodes  = (const int*)  d_in[1];  // [B,C,L] int32
    const float* tree_codes  = (const float*)d_in[2];  // [B,C,L] f32
    const float* emb         = (const float*)d_in[3];  // [N,D] f32
    const float* W           = (const float*)d_in[4];  // [2N,D] f32
    const float* bias        = (const float*)d_in[5];  // [2N] f32

    float* partial = (float*)d_ws;   // B_SZ floats of scratch
    float* out     = (float*)d_out;  // scalar loss

    hsm_fused_loss_kernel<<<B_SZ, 128, 0, stream>>>(
        input_nodes, path_nodes, tree_codes, emb, W, bias, partial);
    hsm_reduce_kernel<<<1, 256, 0, stream>>>(partial, out);
}